// ContextQueryAttention_7284264534761
// MI455X (gfx1250) — compile-verified
//
#include <hip/hip_runtime.h>
#include <hip/hip_bf16.h>

// ---- problem constants (from reference setup_inputs) ----
constexpr int B = 64;
constexpr int H = 128;   // hidden
constexpr int C = 1024;  // context length
constexpr int Q = 128;   // query length
constexpr float NEG_INF = -1e30f;

typedef float v2f __attribute__((ext_vector_type(2)));
typedef float v8f __attribute__((ext_vector_type(8)));

// fp32 WMMA: D(16x16) = A(16x4) * B(4x16) + C, wave32.
__device__ __forceinline__ v8f wmma_f32(v2f a, v2f b, v8f c) {
    return __builtin_amdgcn_wmma_f32_16x16x4_f32(
        /*neg_a=*/false, a, /*neg_b=*/false, b,
        /*c_mod=*/(short)0, c, /*reuse_a=*/false, /*reuse_b=*/false);
}

// ---------------------------------------------------------------------------
// K0: s0[b,c] = sum_h c[b,h,c]*cw[h] ;  s1[b,q] = sum_h q[b,h,q]*qw[h]
// ---------------------------------------------------------------------------
__global__ __launch_bounds__(256)
void k0_proj(const float* __restrict__ cIn, const float* __restrict__ qIn,
             const float* __restrict__ cw, const float* __restrict__ qw,
             float* __restrict__ s0, float* __restrict__ s1) {
    int idx = blockIdx.x * blockDim.x + threadIdx.x;
    if (idx < B * C) {
        int b = idx / C, ci = idx % C;
        const float* p = cIn + (size_t)b * H * C + ci;
        float acc = 0.f;
        #pragma unroll 4
        for (int h = 0; h < H; ++h) acc += p[(size_t)h * C] * cw[h];
        s0[idx] = acc;
    } else if (idx < B * C + B * Q) {
        int j = idx - B * C;
        int b = j / Q, qq = j % Q;
        const float* p = qIn + (size_t)b * H * Q + qq;
        float acc = 0.f;
        #pragma unroll 4
        for (int h = 0; h < H; ++h) acc += p[(size_t)h * Q] * qw[h];
        s1[j] = acc;
    }
}

// ---------------------------------------------------------------------------
// K1: S[b,c,q] = s0[b,c] + s1[b,q] + bias + sum_h (c[b,h,c]*cqw[h]) * q[b,h,q]
// Wave computes a 16(M=c) x 32(N=q) tile: one shared A fragment, 2 WMMAs/step.
// Block = 8 waves covering 2 c-tiles x full Q. Grid = B*(C/32).
// ---------------------------------------------------------------------------
__global__ __launch_bounds__(256)
void k1_sgemm(const float* __restrict__ cIn, const float* __restrict__ qIn,
              const float* __restrict__ cqw, const float* __restrict__ s0,
              const float* __restrict__ s1, const float* __restrict__ bias,
              float* __restrict__ S) {
    int blk = blockIdx.x;              // b*(C/32) + cpair
    int b = blk >> 5;                  // C/32 == 32
    int cpair = blk & 31;
    int wave = threadIdx.x >> 5;
    int lane = threadIdx.x & 31;
    int g = lane >> 4, ln = lane & 15;
    int cbase = (cpair * 2 + (wave >> 2)) * 16;
    int qbase0 = (wave & 3) * 32;
    int qbase1 = qbase0 + 16;

    const float* cB = cIn + (size_t)b * H * C;
    const float* qB = qIn + (size_t)b * H * Q;

    v8f acc0 = {}, acc1 = {};
    for (int k0 = 0; k0 < H; k0 += 4) {
        int ka = k0 + 2 * g;
        v2f A, B0, B1;
        // A[m=ln][k] = c[b,k,cbase+ln] * cqw[k]   (shared by both WMMAs)
        A.x = cB[(size_t)ka * C + cbase + ln] * cqw[ka];
        A.y = cB[(size_t)(ka + 1) * C + cbase + ln] * cqw[ka + 1];
        // B[k][n] = q[b,k,qbase+n]
        B0.x = qB[(size_t)ka * Q + qbase0 + ln];
        B0.y = qB[(size_t)(ka + 1) * Q + qbase0 + ln];
        B1.x = qB[(size_t)ka * Q + qbase1 + ln];
        B1.y = qB[(size_t)(ka + 1) * Q + qbase1 + ln];
        acc0 = wmma_f32(A, B0, acc0);
        acc1 = wmma_f32(A, B1, acc1);
    }

    float bv = bias[0];
    float s1v0 = s1[b * Q + qbase0 + ln] + bv;
    float s1v1 = s1[b * Q + qbase1 + ln] + bv;
    float* Sp0 = S + ((size_t)b * C + cbase) * Q + qbase0 + ln;
    float* Sp1 = S + ((size_t)b * C + cbase) * Q + qbase1 + ln;
    #pragma unroll
    for (int r = 0; r < 8; ++r) {
        int m = r + 8 * g;             // D: lane holds n=ln, VGPR r -> row m
        float s0v = s0[b * C + cbase + m];
        Sp0[(size_t)m * Q] = acc0[r] + s0v + s1v0;
        Sp1[(size_t)m * Q] = acc1[r] + s0v + s1v1;
    }
}

// ---------------------------------------------------------------------------
// K2: per-row (over q) softmax stats: rmax[b,c], rinv[b,c] = 1/sum(exp)
// One block (128 thr) per row.
// ---------------------------------------------------------------------------
__global__ __launch_bounds__(128)
void k2_rowstats(const float* __restrict__ S, const float* __restrict__ q_mask,
                 float* __restrict__ rmax, float* __restrict__ rinv) {
    int row = blockIdx.x;              // b*C + c
    int b = row / C;
    int t = threadIdx.x;               // q index
    float v = S[(size_t)row * Q + t];
    if (q_mask[b * Q + t] <= 0.f) v = NEG_INF;

    __shared__ float red[128];
    red[t] = v; __syncthreads();
    for (int s = 64; s > 0; s >>= 1) {
        if (t < s) red[t] = fmaxf(red[t], red[t + s]);
        __syncthreads();
    }
    float m = red[0]; __syncthreads();
    float e = __expf(v - m);
    red[t] = e; __syncthreads();
    for (int s = 64; s > 0; s >>= 1) {
        if (t < s) red[t] += red[t + s];
        __syncthreads();
    }
    if (t == 0) { rmax[row] = m; rinv[row] = 1.f / red[0]; }
}

// ---------------------------------------------------------------------------
// K3: per-column (over c) softmax stats: cmax[b,q], cinv[b,q]
// Block (16,16): tx = q within tile, ty strides c. Grid = B*(Q/16).
// ---------------------------------------------------------------------------
__global__ __launch_bounds__(256)
void k3_colstats(const float* __restrict__ S, const float* __restrict__ c_mask,
                 float* __restrict__ cmax, float* __restrict__ cinv) {
    int b = blockIdx.x >> 3;           // Q/16 == 8
    int qt = blockIdx.x & 7;
    int tx = threadIdx.x, ty = threadIdx.y;
    int qq = qt * 16 + tx;

    __shared__ float red[16][17];
    const float* Sb = S + (size_t)b * C * Q + qq;
    const float* cm = c_mask + (size_t)b * C;

    float mx = NEG_INF;
    for (int ci = ty; ci < C; ci += 16) {
        float v = Sb[(size_t)ci * Q];
        if (cm[ci] <= 0.f) v = NEG_INF;
        mx = fmaxf(mx, v);
    }
    red[ty][tx] = mx; __syncthreads();
    for (int s = 8; s > 0; s >>= 1) {
        if (ty < s) red[ty][tx] = fmaxf(red[ty][tx], red[ty + s][tx]);
        __syncthreads();
    }
    float m = red[0][tx]; __syncthreads();

    float ps = 0.f;
    for (int ci = ty; ci < C; ci += 16) {
        float v = Sb[(size_t)ci * Q];
        if (cm[ci] <= 0.f) v = NEG_INF;
        ps += __expf(v - m);
    }
    red[ty][tx] = ps; __syncthreads();
    for (int s = 8; s > 0; s >>= 1) {
        if (ty < s) red[ty][tx] += red[ty + s][tx];
        __syncthreads();
    }
    if (ty == 0) { cmax[b * Q + qq] = m; cinv[b * Q + qq] = 1.f / red[0][tx]; }
}

// ---------------------------------------------------------------------------
// K4: T[b,q,h] = sum_c b_att[b,c,q] * c[b,h,c]; b_att computed on the fly.
// Wave computes 16(M=q) x 32(N=h): one exp-A fragment feeds 2 WMMAs/step.
// Block = 8 waves covering 2 q-tiles x full H. Grid = B*(Q/32).
// ---------------------------------------------------------------------------
__global__ __launch_bounds__(256)
void k4_tgemm(const float* __restrict__ S, const float* __restrict__ cIn,
              const float* __restrict__ c_mask, const float* __restrict__ cmax,
              const float* __restrict__ cinv, float* __restrict__ T) {
    int b = blockIdx.x >> 2;           // Q/32 == 4
    int qpair = blockIdx.x & 3;
    int wave = threadIdx.x >> 5;
    int lane = threadIdx.x & 31;
    int g = lane >> 4, ln = lane & 15;
    int qbase = (qpair * 2 + (wave >> 2)) * 16;
    int hbase0 = (wave & 3) * 32;
    int hbase1 = hbase0 + 16;

    // A-fragment row (M = q) is fixed per lane -> load its stats once.
    float cmv = cmax[b * Q + qbase + ln];
    float civ = cinv[b * Q + qbase + ln];
    const float* Sb = S + (size_t)b * C * Q;
    const float* cmk = c_mask + (size_t)b * C;
    const float* cB = cIn + (size_t)b * H * C;

    v8f acc0 = {}, acc1 = {};
    for (int k0 = 0; k0 < C; k0 += 4) {
        int ka = k0 + 2 * g;
        v2f A, B0, B1;
        // A[m=q][k=c] = exp(masked S[c,q] - cmax[q]) * cinv[q]
        float v0 = Sb[(size_t)ka * Q + qbase + ln];
        float v1 = Sb[(size_t)(ka + 1) * Q + qbase + ln];
        if (cmk[ka] <= 0.f)     v0 = NEG_INF;
        if (cmk[ka + 1] <= 0.f) v1 = NEG_INF;
        A.x = __expf(v0 - cmv) * civ;
        A.y = __expf(v1 - cmv) * civ;
        // B[k=c][n=h] = c[b,h,c]: two consecutive c -> one float2 load
        B0 = *reinterpret_cast<const v2f*>(&cB[(size_t)(hbase0 + ln) * C + ka]);
        B1 = *reinterpret_cast<const v2f*>(&cB[(size_t)(hbase1 + ln) * C + ka]);
        acc0 = wmma_f32(A, B0, acc0);
        acc1 = wmma_f32(A, B1, acc1);
    }

    float* Tp0 = T + ((size_t)b * Q + qbase) * H + hbase0 + ln;
    float* Tp1 = T + ((size_t)b * Q + qbase) * H + hbase1 + ln;
    #pragma unroll
    for (int r = 0; r < 8; ++r) {
        int m = r + 8 * g;
        Tp0[(size_t)m * H] = acc0[r];
        Tp1[(size_t)m * H] = acc1[r];
    }
}

// ---------------------------------------------------------------------------
// K5 (fused): a = a_att@q_t, bmat = a_att@T; one exp-A fragment feeds
// 4 WMMAs/step (2 h-tiles x {a,b}). Then out = [c ; a ; c*a ; c*bmat]
// in (B,4H,C) layout. Block = 8 waves covering 2 c-tiles. Grid = B*(C/32).
// ---------------------------------------------------------------------------
__global__ __launch_bounds__(256)
void k5_fused_out(const float* __restrict__ S, const float* __restrict__ qIn,
                  const float* __restrict__ T, const float* __restrict__ cIn,
                  const float* __restrict__ q_mask, const float* __restrict__ rmax,
                  const float* __restrict__ rinv, float* __restrict__ out) {
    int b = blockIdx.x >> 5;           // C/32 == 32
    int cpair = blockIdx.x & 31;
    int wave = threadIdx.x >> 5;
    int lane = threadIdx.x & 31;
    int g = lane >> 4, ln = lane & 15;
    int cbase = (cpair * 2 + (wave >> 2)) * 16;
    int hbase0 = (wave & 3) * 32;
    int hbase1 = hbase0 + 16;

    int crow = cbase + ln;             // A-fragment row (M = c) per lane
    float rmv = rmax[b * C + crow];
    float riv = rinv[b * C + crow];
    const float* Srow = S + ((size_t)b * C + crow) * Q;
    const float* qmk = q_mask + (size_t)b * Q;
    const float* qB = qIn + (size_t)b * H * Q;
    const float* Tb = T + (size_t)b * Q * H;

    v8f accA0 = {}, accA1 = {};   // a (h-tile 0, 1)
    v8f accB0 = {}, accB1 = {};   // b (h-tile 0, 1)
    for (int k0 = 0; k0 < Q; k0 += 4) {
        int ka = k0 + 2 * g;
        // A[m=c][k=q] = a_att on the fly (two consecutive q -> float2 load)
        v2f sv = *reinterpret_cast<const v2f*>(&Srow[ka]);
        float v0 = (qmk[ka] > 0.f)     ? sv.x : NEG_INF;
        float v1 = (qmk[ka + 1] > 0.f) ? sv.y : NEG_INF;
        v2f A;
        A.x = __expf(v0 - rmv) * riv;
        A.y = __expf(v1 - rmv) * riv;
        // B1[k=q][n=h] = q[b,h,q]: consecutive q -> float2 loads
        v2f Bq0 = *reinterpret_cast<const v2f*>(&qB[(size_t)(hbase0 + ln) * Q + ka]);
        v2f Bq1 = *reinterpret_cast<const v2f*>(&qB[(size_t)(hbase1 + ln) * Q + ka]);
        // B2[k=q][n=h] = T[b,q,h]
        v2f Bt0, Bt1;
        Bt0.x = Tb[(size_t)ka * H + hbase0 + ln];
        Bt0.y = Tb[(size_t)(ka + 1) * H + hbase0 + ln];
        Bt1.x = Tb[(size_t)ka * H + hbase1 + ln];
        Bt1.y = Tb[(size_t)(ka + 1) * H + hbase1 + ln];
        accA0 = wmma_f32(A, Bq0, accA0);
        accA1 = wmma_f32(A, Bq1, accA1);
        accB0 = wmma_f32(A, Bt0, accB0);
        accB1 = wmma_f32(A, Bt1, accB1);
    }

    int h0 = hbase0 + ln;              // D: lane holds n=ln
    int h1 = hbase1 + ln;
    const float* cRow0 = cIn + ((size_t)b * H + h0) * C;
    const float* cRow1 = cIn + ((size_t)b * H + h1) * C;
    float* outB = out + (size_t)b * 4 * H * C;
    #pragma unroll
    for (int r = 0; r < 8; ++r) {
        int m = r + 8 * g;
        int cidx = cbase + m;
        float cv0 = cRow0[cidx], cv1 = cRow1[cidx];
        float a0 = accA0[r], a1 = accA1[r];
        float b0 = accB0[r], b1 = accB1[r];
        outB[(size_t)h0 * C + cidx]           = cv0;        // c
        outB[(size_t)(H + h0) * C + cidx]     = a0;         // a
        outB[(size_t)(2 * H + h0) * C + cidx] = cv0 * a0;   // c*a
        outB[(size_t)(3 * H + h0) * C + cidx] = cv0 * b0;   // c*b
        outB[(size_t)h1 * C + cidx]           = cv1;
        outB[(size_t)(H + h1) * C + cidx]     = a1;
        outB[(size_t)(2 * H + h1) * C + cidx] = cv1 * a1;
        outB[(size_t)(3 * H + h1) * C + cidx] = cv1 * b1;
    }
}

// ---------------------------------------------------------------------------
extern "C" void kernel_launch(void* const* d_in, const int* in_sizes, int n_in,
                              void* d_out, int out_size, void* d_ws, size_t ws_size,
                              hipStream_t stream) {
    const float* cIn   = (const float*)d_in[0];  // (B,H,C)
    const float* qIn   = (const float*)d_in[1];  // (B,H,Q)
    const float* cmask = (const float*)d_in[2];  // (B,C)
    const float* qmask = (const float*)d_in[3];  // (B,Q)
    const float* cw    = (const float*)d_in[4];  // (H,1)
    const float* qw    = (const float*)d_in[5];  // (H,1)
    const float* cqw   = (const float*)d_in[6];  // (1,1,H)
    const float* bias  = (const float*)d_in[7];  // (1,)
    float* out = (float*)d_out;                  // (B,4H,C)

    // workspace layout (floats)
    float* ws   = (float*)d_ws;
    float* S    = ws;                              // B*C*Q
    float* rmax = S    + (size_t)B * C * Q;        // B*C
    float* rinv = rmax + (size_t)B * C;            // B*C
    float* cmax = rinv + (size_t)B * C;            // B*Q
    float* cinv = cmax + (size_t)B * Q;            // B*Q
    float* T    = cinv + (size_t)B * Q;            // B*Q*H
    float* s0   = T    + (size_t)B * Q * H;        // B*C
    float* s1   = s0   + (size_t)B * C;            // B*Q

    (void)in_sizes; (void)n_in; (void)out_size; (void)ws_size;

    int n0 = B * C + B * Q;
    k0_proj<<<(n0 + 255) / 256, 256, 0, stream>>>(cIn, qIn, cw, qw, s0, s1);

    k1_sgemm<<<B * (C / 32), 256, 0, stream>>>(cIn, qIn, cqw, s0, s1, bias, S);

    k2_rowstats<<<B * C, 128, 0, stream>>>(S, qmask, rmax, rinv);

    k3_colstats<<<B * (Q / 16), dim3(16, 16), 0, stream>>>(S, cmask, cmax, cinv);

    k4_tgemm<<<B * (Q / 32), 256, 0, stream>>>(S, cIn, cmask, cmax, cinv, T);

    k5_fused_out<<<B * (C / 32), 256, 0, stream>>>(S, qIn, T, cIn, qmask, rmax, rinv, out);
}